// RNadeBase_51170240364858
// MI455X (gfx1250) — compile-verified
//
#include <hip/hip_runtime.h>

typedef __attribute__((ext_vector_type(2))) float v2f;
typedef __attribute__((ext_vector_type(8))) float v8f;

#define Bsz 8
#define Lsz 512
#define NDsz 3
#define CDsz 61
#define Dsz 64
#define Hsz 512
#define Osz 192
#define CHUNKS 16
#define LC (Lsz / CHUNKS)      // 32
#define LDS_STRIDE 516          // 516 % 64 == 4 -> conflict-free wave32 b64 reads

// Workspace layout (floats):
//   xcat    : [L][16][D]      = 524288
//   u / ar  : [L][8][H]       = 2097152   (u from encode GEMM, rewritten in place to relu(a))
//   partial : [CHUNKS][8][H]  = 65536
#define XCAT_ELEMS (Lsz * 16 * Dsz)
#define U_ELEMS    (Lsz * Bsz * Hsz)

// ---------------------------------------------------------------------------
// Phase 0: pack xcat[l][m][d] = concat(x,z) transposed to (L, 16, D), rows 8..15 = 0
// ---------------------------------------------------------------------------
__global__ void pack_xcat(const float* __restrict__ x, const float* __restrict__ z,
                          float* __restrict__ xcat) {
    int idx = blockIdx.x * blockDim.x + threadIdx.x;
    if (idx >= XCAT_ELEMS) return;
    int l = idx >> 10;          // / (16*64)
    int r = idx & 1023;
    int m = r >> 6;
    int d = r & 63;
    float v = 0.f;
    if (m < Bsz) {
        v = (d < NDsz) ? x[(m * Lsz + l) * NDsz + d]
                       : z[(m * Lsz + l) * CDsz + (d - NDsz)];
    }
    xcat[idx] = v;
}

// ---------------------------------------------------------------------------
// Phase 1: u[l] = xcat[l] @ W_enc[l]   (16x512x64 per l, one wave per (l, 16-wide N tile))
// W_enc is read exactly once -> non-temporal loads (don't pollute L2).
// ---------------------------------------------------------------------------
__global__ __launch_bounds__(32) void enc_gemm(const float* __restrict__ xcat,
                                               const float* __restrict__ W_enc,
                                               float* __restrict__ u) {
    int l    = blockIdx.x >> 5;      // H/16 = 32 tiles
    int nt   = blockIdx.x & 31;
    int lane = threadIdx.x;
    int m16  = lane & 15;
    int half = lane >> 4;
    int n    = nt * 16 + m16;        // for B/D frags, lane&15 is the N index

    const float* xa = xcat + (size_t)l * (16 * Dsz);
    const float* wb = W_enc + (size_t)l * Dsz * Hsz;

    v8f acc = {0.f, 0.f, 0.f, 0.f, 0.f, 0.f, 0.f, 0.f};
    #pragma unroll
    for (int kk = 0; kk < Dsz / 4; ++kk) {
        int k0 = 4 * kk + 2 * half;
        v2f av = *reinterpret_cast<const v2f*>(xa + m16 * Dsz + k0);   // A: lane=M (L2-resident)
        v2f bv;
        bv.x = __builtin_nontemporal_load(wb + (size_t)k0 * Hsz + n);        // B: lane=N (stream)
        bv.y = __builtin_nontemporal_load(wb + (size_t)(k0 + 1) * Hsz + n);
        acc = __builtin_amdgcn_wmma_f32_16x16x4_f32(false, av, false, bv,
                                                    (short)0, acc, false, false);
    }
    // D layout: lanes 0..15 hold M = vgpr index (0..7) = batch rows; rows 8..15 discarded
    if (lane < 16) {
        float* uo = u + (size_t)l * (Bsz * Hsz) + n;
        #pragma unroll
        for (int v = 0; v < 8; ++v) uo[(size_t)v * Hsz] = acc[v];
    }
}

// ---------------------------------------------------------------------------
// Phase 2a: per-chunk sums of u over L
// ---------------------------------------------------------------------------
__global__ void chunk_sum(const float* __restrict__ u, float* __restrict__ partial) {
    int c = blockIdx.x >> 3;
    int b = blockIdx.x & 7;
    int h = threadIdx.x;
    const float* up = u + (size_t)b * Hsz + h;
    float s = 0.f;
    #pragma unroll 4
    for (int i = 0; i < LC; ++i) {
        int l = c * LC + i;
        s += up[(size_t)l * (Bsz * Hsz)];
    }
    partial[((size_t)c * Bsz + b) * Hsz + h] = s;
}

// ---------------------------------------------------------------------------
// Phase 2b: exclusive scan + a_l = r[l]*(b_enc/r[0] + prefix) , ReLU, in place
// (telescoped rescaling: a_l = rescaling[l] * (b_enc/rescaling[0] + sum_{j<l} u_j))
// ---------------------------------------------------------------------------
__global__ void scan_relu(float* __restrict__ u, const float* __restrict__ partial,
                          const float* __restrict__ b_enc, const float* __restrict__ resc) {
    int c = blockIdx.x >> 3;
    int b = blockIdx.x & 7;
    int h = threadIdx.x;

    float acc = 0.f;
    for (int cp = 0; cp < c; ++cp)
        acc += partial[((size_t)cp * Bsz + b) * Hsz + h];

    float bs = b_enc[h] / resc[0];
    float* up = u + (size_t)b * Hsz + h;
    for (int i = 0; i < LC; ++i) {
        int l = c * LC + i;
        size_t off = (size_t)l * (Bsz * Hsz);
        float uv = up[off];                       // u_l (read before overwrite)
        float a  = resc[l] * (bs + acc);          // exclusive prefix -> a_l
        up[off]  = fmaxf(a, 0.f);                 // store relu(a_l)
        acc += uv;
    }
}

// ---------------------------------------------------------------------------
// Phase 3: out[:, l, :] = relu(a_l) @ W_out[l] + b_out[l]
// One 384-thread workgroup (12 waves) per l; A tile staged in LDS (padded to
// 16 rows, stride 516 -> conflict-free ds_load_b64); each wave owns one N tile.
// W_out is read exactly once -> non-temporal loads; out written once -> NT store.
// ---------------------------------------------------------------------------
__global__ __launch_bounds__(384) void out_gemm(const float* __restrict__ ar,
                                                const float* __restrict__ W_out,
                                                const float* __restrict__ b_out,
                                                float* __restrict__ out) {
    __shared__ float lds[16 * LDS_STRIDE];
    int l   = blockIdx.x;
    int tid = threadIdx.x;

    for (int t = tid; t < 16 * Hsz; t += 384) {
        int m = t >> 9;
        int k = t & 511;
        lds[m * LDS_STRIDE + k] = (m < Bsz) ? ar[(size_t)l * (Bsz * Hsz) + t] : 0.f;
    }
    __syncthreads();

    int w    = tid >> 5;            // 12 waves -> 12 N tiles (O = 192)
    int lane = tid & 31;
    int m16  = lane & 15;
    int half = lane >> 4;
    int n    = w * 16 + m16;

    const float* wb = W_out + (size_t)l * Hsz * Osz;

    v8f acc = {0.f, 0.f, 0.f, 0.f, 0.f, 0.f, 0.f, 0.f};
    #pragma unroll 4
    for (int kk = 0; kk < Hsz / 4; ++kk) {
        int k0 = 4 * kk + 2 * half;
        v2f av = *reinterpret_cast<const v2f*>(&lds[m16 * LDS_STRIDE + k0]);  // ds_load_b64
        v2f bv;
        bv.x = __builtin_nontemporal_load(wb + (size_t)k0 * Osz + n);
        bv.y = __builtin_nontemporal_load(wb + (size_t)(k0 + 1) * Osz + n);
        acc = __builtin_amdgcn_wmma_f32_16x16x4_f32(false, av, false, bv,
                                                    (short)0, acc, false, false);
    }

    if (lane < 16) {
        float bo = b_out[(size_t)l * Osz + n];
        #pragma unroll
        for (int v = 0; v < 8; ++v)
            __builtin_nontemporal_store(acc[v] + bo,
                out + (size_t)v * Lsz * Osz + (size_t)l * Osz + n);
    }
}

// ---------------------------------------------------------------------------
extern "C" void kernel_launch(void* const* d_in, const int* in_sizes, int n_in,
                              void* d_out, int out_size, void* d_ws, size_t ws_size,
                              hipStream_t stream) {
    const float* x     = (const float*)d_in[0];
    const float* z     = (const float*)d_in[1];
    const float* W_enc = (const float*)d_in[2];
    const float* b_enc = (const float*)d_in[3];
    const float* W_out = (const float*)d_in[4];
    const float* b_out = (const float*)d_in[5];
    const float* resc  = (const float*)d_in[6];
    float* out = (float*)d_out;

    float* ws      = (float*)d_ws;
    float* xcat    = ws;                          // XCAT_ELEMS
    float* u       = ws + XCAT_ELEMS;             // U_ELEMS (becomes relu(a) in place)
    float* partial = u + U_ELEMS;                 // CHUNKS*Bsz*Hsz

    pack_xcat<<<(XCAT_ELEMS + 255) / 256, 256, 0, stream>>>(x, z, xcat);
    enc_gemm <<<Lsz * (Hsz / 16), 32, 0, stream>>>(xcat, W_enc, u);
    chunk_sum<<<CHUNKS * Bsz, Hsz, 0, stream>>>(u, partial);
    scan_relu<<<CHUNKS * Bsz, Hsz, 0, stream>>>(u, partial, b_enc, resc);
    out_gemm <<<Lsz, 384, 0, stream>>>(u, W_out, b_out, out);
}